// SSMblock_57294863729228
// MI455X (gfx1250) — compile-verified
//
#include <hip/hip_runtime.h>
#include <math.h>

typedef __attribute__((ext_vector_type(2))) float v2f;
typedef __attribute__((ext_vector_type(8))) float v8f;

#define P_DIM 128
#define H_DIM 256
#define B_DIM 8
#define L_DIM 4096
#define NC    64   // number of chunks
#define CN    64   // chunk length; NC*CN == L_DIM

// ---------------------------------------------------------------------------
// Kernel 1: precompute Lam_ = exp(Lam*delta) and B_ = M @ Bt using the
// closed form B_[p,h] = S[h] - Bt[p,h]/Lam[p],  S[h] = sum_j (Lam_[j]/Lam[j])Bt[j,h]
// ---------------------------------------------------------------------------
__global__ __launch_bounds__(256) void ssm_precompute(
    const float* __restrict__ Lre, const float* __restrict__ Lim,
    const float* __restrict__ Btre, const float* __restrict__ Btim,
    const float* __restrict__ delta,
    float* __restrict__ lam_re, float* __restrict__ lam_im,
    float* __restrict__ Bre, float* __restrict__ Bim) {
  __shared__ float w_re[P_DIM], w_im[P_DIM], iv_re[P_DIM], iv_im[P_DIM];
  const int t = threadIdx.x;
  if (t < P_DIM) {
    float lr = Lre[t], li = Lim[t], d = delta[t];
    float er = __expf(lr * d);
    float ar = er * __cosf(li * d);
    float ai = er * __sinf(li * d);
    lam_re[t] = ar; lam_im[t] = ai;
    float den = lr * lr + li * li;
    float inv_r = lr / den, inv_i = -li / den;        // 1/Lam (conj/|.|^2)
    w_re[t] = ar * inv_r - ai * inv_i;                // Lam_/Lam
    w_im[t] = ar * inv_i + ai * inv_r;
    iv_re[t] = inv_r; iv_im[t] = inv_i;
  }
  __syncthreads();
  const int h = t;                                    // 0..255 covers H
  float s_re = 0.f, s_im = 0.f;
  for (int p = 0; p < P_DIM; ++p) {
    float br = Btre[p * H_DIM + h], bi = Btim[p * H_DIM + h];
    s_re += w_re[p] * br - w_im[p] * bi;
    s_im += w_re[p] * bi + w_im[p] * br;
  }
  for (int p = 0; p < P_DIM; ++p) {
    float br = Btre[p * H_DIM + h], bi = Btim[p * H_DIM + h];
    float qr = iv_re[p] * br - iv_im[p] * bi;
    float qi = iv_re[p] * bi + iv_im[p] * br;
    Bre[p * H_DIM + h] = s_re - qr;
    Bim[p * H_DIM + h] = s_im - qi;
  }
}

// ---------------------------------------------------------------------------
// Kernel 2: Bu[l,b,p] = sum_h x[b,l,h] * B_[p,h]   (re and im GEMMs, WMMA f32)
// Rows r = l*B + b (M=32768), cols = p (N=128), K = H.
// Block: 256 thr = 8 waves, wave w owns p-tile [16w,16w+16), block owns 16 rows.
// ---------------------------------------------------------------------------
__global__ __launch_bounds__(256) void ssm_gemm_bu(
    const float* __restrict__ x, const float* __restrict__ Bre,
    const float* __restrict__ Bim,
    float* __restrict__ bu_re, float* __restrict__ bu_im) {
  const int lane = threadIdx.x & 31;
  const int wave = threadIdx.x >> 5;
  const int m0 = blockIdx.x * 16;
  const int n0 = wave * 16;
  const int lN = lane & 15;
  const int lh = lane >> 4;            // lane half: selects K pair / M half
  const int r = m0 + lN;               // A-matrix row for this lane
  const int b = r & 7, l = r >> 3;
  const float* xrow = x + (size_t)b * L_DIM * H_DIM + (size_t)l * H_DIM + 2 * lh;
  const float* wre = Bre + (size_t)(n0 + lN) * H_DIM + 2 * lh;
  const float* wim = Bim + (size_t)(n0 + lN) * H_DIM + 2 * lh;
  v8f acc_re = {}; v8f acc_im = {};
  for (int k0 = 0; k0 < H_DIM; k0 += 4) {
    v2f a  = *(const v2f*)(xrow + k0);
    v2f br = *(const v2f*)(wre + k0);
    v2f bi = *(const v2f*)(wim + k0);
    acc_re = __builtin_amdgcn_wmma_f32_16x16x4_f32(false, a, false, br, (short)0, acc_re, false, false);
    acc_im = __builtin_amdgcn_wmma_f32_16x16x4_f32(false, a, false, bi, (short)0, acc_im, false, false);
  }
  const int pcol = n0 + lN;
#pragma unroll
  for (int v = 0; v < 8; ++v) {
    int rr = m0 + v + 8 * lh;
    size_t o = (size_t)rr * P_DIM + pcol;
    bu_re[o] = acc_re[v];
    bu_im[o] = acc_im[v];
  }
}

// ---------------------------------------------------------------------------
// Kernels 3-5: chunked linear scan  s[l] = Lam_ * s[l-1] + Bu[l]
// ---------------------------------------------------------------------------
__global__ __launch_bounds__(128) void ssm_scan_a(
    const float* __restrict__ bu_re, const float* __restrict__ bu_im,
    const float* __restrict__ lam_re, const float* __restrict__ lam_im,
    float* __restrict__ E_re, float* __restrict__ E_im) {
  const int p = threadIdx.x, c = blockIdx.x, b = blockIdx.y;
  const float ar = lam_re[p], ai = lam_im[p];
  float sr = 0.f, si = 0.f;
  const int l0 = c * CN;
  for (int t = 0; t < CN; ++t) {
    size_t o = ((size_t)(l0 + t) * B_DIM + b) * P_DIM + p;
    float ur = bu_re[o], ui = bu_im[o];
    float nr = ar * sr - ai * si + ur;
    float ni = ar * si + ai * sr + ui;
    sr = nr; si = ni;
  }
  size_t e = ((size_t)c * B_DIM + b) * P_DIM + p;
  E_re[e] = sr; E_im[e] = si;
}

__global__ __launch_bounds__(128) void ssm_scan_b(
    const float* __restrict__ E_re, const float* __restrict__ E_im,
    const float* __restrict__ lam_re, const float* __restrict__ lam_im,
    float* __restrict__ K_re, float* __restrict__ K_im) {
  const int idx = blockIdx.x * 128 + threadIdx.x;   // b*P + p, 0..1023
  const int p = idx & (P_DIM - 1);
  float pr = lam_re[p], pi = lam_im[p];
  for (int i = 0; i < 6; ++i) {                     // a^64 by squaring
    float nr = pr * pr - pi * pi;
    float ni = 2.f * pr * pi;
    pr = nr; pi = ni;
  }
  float kr = 0.f, ki = 0.f;
  for (int c = 0; c < NC; ++c) {
    size_t o = (size_t)c * (B_DIM * P_DIM) + idx;
    K_re[o] = kr; K_im[o] = ki;
    float er = E_re[o], ei = E_im[o];
    float nr = pr * kr - pi * ki + er;
    float ni = pr * ki + pi * kr + ei;
    kr = nr; ki = ni;
  }
}

__global__ __launch_bounds__(128) void ssm_scan_c(
    float* __restrict__ bu_re, float* __restrict__ bu_im,
    const float* __restrict__ lam_re, const float* __restrict__ lam_im,
    const float* __restrict__ K_re, const float* __restrict__ K_im) {
  const int p = threadIdx.x, c = blockIdx.x, b = blockIdx.y;
  const float ar = lam_re[p], ai = lam_im[p];
  size_t e = ((size_t)c * B_DIM + b) * P_DIM + p;
  float sr = K_re[e], si = K_im[e];
  const int l0 = c * CN;
  for (int t = 0; t < CN; ++t) {
    size_t o = ((size_t)(l0 + t) * B_DIM + b) * P_DIM + p;
    float ur = bu_re[o], ui = bu_im[o];
    float nr = ar * sr - ai * si + ur;
    float ni = ar * si + ai * sr + ui;
    sr = nr; si = ni;
    bu_re[o] = sr; bu_im[o] = si;   // states overwrite Bu in place
  }
}

// ---------------------------------------------------------------------------
// Kernel 6: y = s_re@Ct_re^T - s_im@Ct_im^T + x@D_re^T, then exact GeLU.
// (D_im contributes only to the imaginary part since x is real.)
// ---------------------------------------------------------------------------
__global__ __launch_bounds__(256) void ssm_gemm_out(
    const float* __restrict__ s_re, const float* __restrict__ s_im,
    const float* __restrict__ x,
    const float* __restrict__ Ctre, const float* __restrict__ Ctim,
    const float* __restrict__ Dre, float* __restrict__ out) {
  const int lane = threadIdx.x & 31;
  const int wave = threadIdx.x >> 5;
  const int m0 = blockIdx.x * 16;
  const int n0 = (blockIdx.y * 8 + wave) * 16;      // h tile
  const int lN = lane & 15;
  const int lh = lane >> 4;
  const int r = m0 + lN;
  const int b = r & 7, l = r >> 3;
  const float* srow_re = s_re + (size_t)r * P_DIM + 2 * lh;
  const float* srow_im = s_im + (size_t)r * P_DIM + 2 * lh;
  const float* xrow = x + (size_t)b * L_DIM * H_DIM + (size_t)l * H_DIM + 2 * lh;
  const int hrow = n0 + lN;
  const float* ctr = Ctre + (size_t)hrow * P_DIM + 2 * lh;
  const float* cti = Ctim + (size_t)hrow * P_DIM + 2 * lh;
  const float* drw = Dre + (size_t)hrow * H_DIM + 2 * lh;
  v8f acc = {};
  for (int k0 = 0; k0 < P_DIM; k0 += 4) {           // + s_re @ Ct_re^T
    v2f a = *(const v2f*)(srow_re + k0);
    v2f w = *(const v2f*)(ctr + k0);
    acc = __builtin_amdgcn_wmma_f32_16x16x4_f32(false, a, false, w, (short)0, acc, false, false);
  }
  for (int k0 = 0; k0 < P_DIM; k0 += 4) {           // - s_im @ Ct_im^T
    v2f a = *(const v2f*)(srow_im + k0);
    a = -a;                                         // f32 WMMA has no A-neg modifier
    v2f w = *(const v2f*)(cti + k0);
    acc = __builtin_amdgcn_wmma_f32_16x16x4_f32(false, a, false, w, (short)0, acc, false, false);
  }
  for (int k0 = 0; k0 < H_DIM; k0 += 4) {           // + x @ D_re^T
    v2f a = *(const v2f*)(xrow + k0);
    v2f w = *(const v2f*)(drw + k0);
    acc = __builtin_amdgcn_wmma_f32_16x16x4_f32(false, a, false, w, (short)0, acc, false, false);
  }
#pragma unroll
  for (int v = 0; v < 8; ++v) {
    int rr = m0 + v + 8 * lh;
    int bb = rr & 7, ll = rr >> 3;
    float y = acc[v];
    float g = 0.5f * y * (1.0f + erff(y * 0.70710678118654752f));  // exact GeLU
    out[(size_t)bb * L_DIM * H_DIM + (size_t)ll * H_DIM + hrow] = g;
  }
}

// ---------------------------------------------------------------------------
extern "C" void kernel_launch(void* const* d_in, const int* in_sizes, int n_in,
                              void* d_out, int out_size, void* d_ws, size_t ws_size,
                              hipStream_t stream) {
  const float* x     = (const float*)d_in[0];
  const float* Lre   = (const float*)d_in[1];
  const float* Lim   = (const float*)d_in[2];
  const float* Btre  = (const float*)d_in[3];
  const float* Btim  = (const float*)d_in[4];
  const float* Ctre  = (const float*)d_in[5];
  const float* Ctim  = (const float*)d_in[6];
  const float* Dre   = (const float*)d_in[7];
  // d_in[8] = D_im: does not affect real part of output (x is real)
  const float* delta = (const float*)d_in[9];

  float* ws = (float*)d_ws;
  float* lam_re = ws;
  float* lam_im = lam_re + P_DIM;
  float* Bre    = lam_im + P_DIM;
  float* Bim    = Bre + P_DIM * H_DIM;
  float* bu_re  = Bim + P_DIM * H_DIM;
  float* bu_im  = bu_re + (size_t)L_DIM * B_DIM * P_DIM;
  float* E_re   = bu_im + (size_t)L_DIM * B_DIM * P_DIM;
  float* E_im   = E_re + NC * B_DIM * P_DIM;
  float* K_re   = E_im + NC * B_DIM * P_DIM;
  float* K_im   = K_re + NC * B_DIM * P_DIM;

  ssm_precompute<<<1, 256, 0, stream>>>(Lre, Lim, Btre, Btim, delta,
                                        lam_re, lam_im, Bre, Bim);
  ssm_gemm_bu<<<(B_DIM * L_DIM) / 16, 256, 0, stream>>>(x, Bre, Bim, bu_re, bu_im);
  ssm_scan_a<<<dim3(NC, B_DIM), 128, 0, stream>>>(bu_re, bu_im, lam_re, lam_im, E_re, E_im);
  ssm_scan_b<<<(B_DIM * P_DIM) / 128, 128, 0, stream>>>(E_re, E_im, lam_re, lam_im, K_re, K_im);
  ssm_scan_c<<<dim3(NC, B_DIM), 128, 0, stream>>>(bu_re, bu_im, lam_re, lam_im, K_re, K_im);
  ssm_gemm_out<<<dim3((B_DIM * L_DIM) / 16, 2), 256, 0, stream>>>(
      bu_re, bu_im, x, Ctre, Ctim, Dre, (float*)d_out);
}